// WeightedRGCN_2920577761369
// MI455X (gfx1250) — compile-verified
//
#include <hip/hip_runtime.h>

typedef float v2f __attribute__((ext_vector_type(2)));
typedef float v8f __attribute__((ext_vector_type(8)));

constexpr int kUsers  = 200000;
constexpr int kItems  = 100000;
constexpr int kInDim  = 64;
constexpr int kHidden = 128;
constexpr int kOut    = 64;
constexpr int kRel    = 5;
constexpr int kEdges  = 500000;

constexpr int kWaves = 4;            // waves per block in GEMM kernels
constexpr int kBlk   = kWaves * 32;  // wave32

// meta[] layout (ints): [0..4]=hist  [5..10]=relOff  [11..16]=grpOff  [17..21]=cursor
constexpr int M_HIST = 0, M_RELOFF = 5, M_GRPOFF = 11, M_CUR = 17;

// ---------------------------------------------------------------- utility ---

__global__ void rgcn_zero_kernel(float* cntUser, float* cntItem, int* meta) {
  int i = blockIdx.x * blockDim.x + threadIdx.x;
  if (i < kUsers * kRel) cntUser[i] = 0.f;
  if (i < kItems * kRel) cntItem[i] = 0.f;
  if (i < 32) meta[i] = 0;
}

__global__ void rgcn_count_kernel(const int* __restrict__ src, const int* __restrict__ dst,
                                  const int* __restrict__ et,
                                  float* cntUser, float* cntItem, int* meta) {
  int e = blockIdx.x * blockDim.x + threadIdx.x;
  if (e >= kEdges) return;
  int r = et[e];
  atomicAdd(&meta[M_HIST + r], 1);
  atomicAdd(&cntUser[(size_t)src[e] * kRel + r], 1.0f);
  atomicAdd(&cntItem[(size_t)dst[e] * kRel + r], 1.0f);
}

__global__ void rgcn_prefix_kernel(int* meta) {
  if (threadIdx.x != 0 || blockIdx.x != 0) return;
  int ro = 0, go = 0;
  meta[M_RELOFF] = 0;
  meta[M_GRPOFF] = 0;
  for (int r = 0; r < kRel; ++r) {
    meta[M_CUR + r] = ro;          // scatter cursor starts at relation offset
    int h = meta[M_HIST + r];
    ro += h;
    go += (h + 15) >> 4;           // 16-edge WMMA groups per relation
    meta[M_RELOFF + r + 1] = ro;
    meta[M_GRPOFF + r + 1] = go;
  }
}

__global__ void rgcn_bucket_kernel(const int* __restrict__ et, int* meta, int* bucket) {
  int e = blockIdx.x * blockDim.x + threadIdx.x;
  if (e >= kEdges) return;
  int pos = atomicAdd(&meta[M_CUR + et[e]], 1);
  bucket[pos] = e;
}

__global__ void rgcn_relu_kernel(float* p, int n) {
  int i = blockIdx.x * blockDim.x + threadIdx.x;
  if (i < n) p[i] = fmaxf(p[i], 0.f);
}

// ------------------------------------------------- root GEMM: x@Wroot + b ---
// One wave computes a 16-row x DOUT strip with V_WMMA_F32_16X16X4_F32.
// Writes EVERY element of out (also serves as the accumulator init for the
// subsequent atomic edge-message scatter).

template <int DIN, int DOUT>
__global__ void __launch_bounds__(kBlk)
rgcn_root_gemm_kernel(const float* __restrict__ x, const float* __restrict__ W,
                      const float* __restrict__ bias, float* __restrict__ out, int n) {
  __shared__ float ldsA[kWaves][16 * DIN];
  const int lane = threadIdx.x & 31;
  const int wid  = threadIdx.x >> 5;
  const int row0 = (blockIdx.x * kWaves + wid) * 16;
  if (row0 >= n) return;  // wave-uniform: EXEC stays all-ones for WMMA

  const int row  = lane & 15;
  const int half = lane >> 4;
  float* A = ldsA[wid];

  {  // stage 16xDIN A tile; clamp OOB rows (their stores are masked below)
    int srcRow = row0 + row;
    if (srcRow >= n) srcRow = n - 1;
    const float* xp = x + (size_t)srcRow * DIN;
    const int c0 = half * (DIN / 2);
#pragma unroll
    for (int c = 0; c < DIN / 2; ++c) A[row * DIN + c0 + c] = xp[c0 + c];
  }

  const int off = half * 2;  // ISA A layout: lanes16-31 hold K=kc+2,kc+3
#pragma unroll
  for (int ct = 0; ct < DOUT / 16; ++ct) {
    const int col = ct * 16 + row;  // N index = lane%16
    v8f acc;
    const float bv = bias[col];
#pragma unroll
    for (int j = 0; j < 8; ++j) acc[j] = bv;
    const float* Wc = W + col;
#pragma unroll
    for (int kc = 0; kc < DIN; kc += 4) {
      v2f a, b;
      a.x = A[row * DIN + kc + off];
      a.y = A[row * DIN + kc + off + 1];
      b.x = Wc[(size_t)(kc + off) * DOUT];
      b.y = Wc[(size_t)(kc + off + 1) * DOUT];
      acc = __builtin_amdgcn_wmma_f32_16x16x4_f32(false, a, false, b, (short)0, acc,
                                                  false, false);
    }
#pragma unroll
    for (int j = 0; j < 8; ++j) {          // C/D layout: M = j + 8*half
      const int rr = row0 + j + half * 8;
      if (rr < n) out[(size_t)rr * DOUT + col] = acc[j];
    }
  }
}

// ------------------------------------- edge messages: out[dst] += mean msg ---
// Edges are bucketed by relation, so each wave's 16-edge group shares one W_r.
// A rows are gathered source features pre-scaled by 1/max(cnt(dst,rel),1):
// scatter-adding the WMMA result then realizes   sum_r mean_r(x_src @ W_r).

template <int DIN, int DOUT>
__global__ void __launch_bounds__(kBlk)
rgcn_edge_msg_kernel(const float* __restrict__ xsrc, const float* __restrict__ Wrel,
                     const int* __restrict__ gidx, const int* __restrict__ sidx,
                     const float* __restrict__ cnt, const int* __restrict__ meta,
                     const int* __restrict__ bucket, float* __restrict__ out) {
  __shared__ float ldsA[kWaves][16 * DIN];
  const int lane = threadIdx.x & 31;
  const int wid  = threadIdx.x >> 5;
  const int grp  = blockIdx.x * kWaves + wid;
  if (grp >= meta[M_GRPOFF + kRel]) return;  // wave-uniform exit

  int r = 0;  // locate this group's relation bucket
  while (r < kRel - 1 && grp >= meta[M_GRPOFF + r + 1]) ++r;
  const int e0 = meta[M_RELOFF + r] + (grp - meta[M_GRPOFF + r]) * 16;
  int m = meta[M_RELOFF + r + 1] - e0;
  if (m > 16) m = 16;                        // tail group: <16 live edges

  int g = 0, s = 0;
  float scl = 0.f;                           // dead rows contribute zeros
  if (lane < m) {                            // lanes 0..15 fetch edge metadata
    const int eid = bucket[e0 + lane];
    g   = gidx[eid];
    s   = sidx[eid];
    scl = 1.0f / fmaxf(cnt[(size_t)s * kRel + r], 1.0f);
  }

  const int row  = lane & 15;
  const int half = lane >> 4;
  float* A = ldsA[wid];

  {  // gather + pre-scale the 16xDIN A tile
    const int   gRow  = __shfl(g, row, 32);
    const float scRow = __shfl(scl, row, 32);
    const float* xp = xsrc + (size_t)gRow * DIN;
    const int c0 = half * (DIN / 2);
#pragma unroll
    for (int c = 0; c < DIN / 2; ++c) A[row * DIN + c0 + c] = xp[c0 + c] * scRow;
  }

  const int off = half * 2;
#pragma unroll
  for (int ct = 0; ct < DOUT / 16; ++ct) {
    const int col = ct * 16 + row;
    v8f acc;
#pragma unroll
    for (int j = 0; j < 8; ++j) acc[j] = 0.f;
    const float* Wc = Wrel + (size_t)r * DIN * DOUT + col;
#pragma unroll
    for (int kc = 0; kc < DIN; kc += 4) {
      v2f a, b;
      a.x = A[row * DIN + kc + off];
      a.y = A[row * DIN + kc + off + 1];
      b.x = Wc[(size_t)(kc + off) * DOUT];
      b.y = Wc[(size_t)(kc + off + 1) * DOUT];
      acc = __builtin_amdgcn_wmma_f32_16x16x4_f32(false, a, false, b, (short)0, acc,
                                                  false, false);
    }
#pragma unroll
    for (int j = 0; j < 8; ++j) {
      const int slot = j + half * 8;         // edge slot of this acc element
      if (slot < m) {
        const int sRow = __shfl(s, slot, 32);
        atomicAdd(out + (size_t)sRow * DOUT + col, acc[j]);
      }
    }
  }
}

// ------------------------------------------------------------------ driver ---

extern "C" void kernel_launch(void* const* d_in, const int* in_sizes, int n_in,
                              void* d_out, int out_size, void* d_ws, size_t ws_size,
                              hipStream_t stream) {
  (void)in_sizes; (void)n_in; (void)out_size; (void)ws_size;

  const float* x_user   = (const float*)d_in[0];
  const float* x_item   = (const float*)d_in[1];
  const int*   edge_ix  = (const int*)d_in[2];
  const int*   et       = (const int*)d_in[3];
  const float* W1_ui    = (const float*)d_in[4];
  const float* root1_ui = (const float*)d_in[5];
  const float* b1_ui    = (const float*)d_in[6];
  const float* W1_iu    = (const float*)d_in[7];
  const float* root1_iu = (const float*)d_in[8];
  const float* b1_iu    = (const float*)d_in[9];
  const float* W2_ui    = (const float*)d_in[10];
  const float* root2_ui = (const float*)d_in[11];
  const float* b2_ui    = (const float*)d_in[12];
  const float* W2_iu    = (const float*)d_in[13];
  const float* root2_iu = (const float*)d_in[14];
  const float* b2_iu    = (const float*)d_in[15];

  const int* src = edge_ix;            // users
  const int* dst = edge_ix + kEdges;   // items

  // workspace carve-up (~162 MB)
  char* ws = (char*)d_ws;
  size_t o = 0;
  auto take = [&](size_t bytes) -> char* {
    char* p = ws + o;
    o += (bytes + 255) & ~(size_t)255;
    return p;
  };
  float* cntItem = (float*)take((size_t)kItems * kRel * 4);
  float* cntUser = (float*)take((size_t)kUsers * kRel * 4);
  int*   meta    = (int*)take(32 * 4);
  int*   bucket  = (int*)take((size_t)kEdges * 4);
  float* item1   = (float*)take((size_t)kItems * kHidden * 4);
  float* user1   = (float*)take((size_t)kUsers * kHidden * 4);

  // ---- graph preprocessing (counts + relation buckets) ----
  const int zt = kUsers * kRel;
  rgcn_zero_kernel<<<(zt + 255) / 256, 256, 0, stream>>>(cntUser, cntItem, meta);
  rgcn_count_kernel<<<(kEdges + 255) / 256, 256, 0, stream>>>(src, dst, et,
                                                              cntUser, cntItem, meta);
  rgcn_prefix_kernel<<<1, 32, 0, stream>>>(meta);
  rgcn_bucket_kernel<<<(kEdges + 255) / 256, 256, 0, stream>>>(et, meta, bucket);

  const int gmax    = (kEdges + 15) / 16 + kRel;          // >= total 16-edge groups
  const int eblocks = (gmax + kWaves - 1) / kWaves;
  auto gblocks = [](int n) { return ((n + 15) / 16 + kWaves - 1) / kWaves; };

  // ---- layer 1: item1 = relu(rgcn(user->item)), user1 = relu(rgcn(item->user))
  rgcn_root_gemm_kernel<kInDim, kHidden><<<gblocks(kItems), kBlk, 0, stream>>>(
      x_item, root1_ui, b1_ui, item1, kItems);
  rgcn_edge_msg_kernel<kInDim, kHidden><<<eblocks, kBlk, 0, stream>>>(
      x_user, W1_ui, src, dst, cntItem, meta, bucket, item1);
  rgcn_relu_kernel<<<(kItems * kHidden + 255) / 256, 256, 0, stream>>>(
      item1, kItems * kHidden);

  rgcn_root_gemm_kernel<kInDim, kHidden><<<gblocks(kUsers), kBlk, 0, stream>>>(
      x_user, root1_iu, b1_iu, user1, kUsers);
  rgcn_edge_msg_kernel<kInDim, kHidden><<<eblocks, kBlk, 0, stream>>>(
      x_item, W1_iu, dst, src, cntUser, meta, bucket, user1);
  rgcn_relu_kernel<<<(kUsers * kHidden + 255) / 256, 256, 0, stream>>>(
      user1, kUsers * kHidden);

  // ---- layer 2 -> d_out: rows [0,kUsers) = user2, rows [kUsers,..) = item2
  float* user2 = (float*)d_out;
  float* item2 = (float*)d_out + (size_t)kUsers * kOut;

  rgcn_root_gemm_kernel<kHidden, kOut><<<gblocks(kUsers), kBlk, 0, stream>>>(
      user1, root2_iu, b2_iu, user2, kUsers);
  rgcn_edge_msg_kernel<kHidden, kOut><<<eblocks, kBlk, 0, stream>>>(
      item1, W2_iu, dst, src, cntUser, meta, bucket, user2);

  rgcn_root_gemm_kernel<kHidden, kOut><<<gblocks(kItems), kBlk, 0, stream>>>(
      item1, root2_ui, b2_ui, item2, kItems);
  rgcn_edge_msg_kernel<kHidden, kOut><<<eblocks, kBlk, 0, stream>>>(
      user1, W2_ui, src, dst, cntItem, meta, bucket, item2);
}